// DeepMDsimpleEnergy_481036337906
// MI455X (gfx1250) — compile-verified
//
#include <hip/hip_runtime.h>
#include <math.h>

// ---------------------------------------------------------------------------
// DeepMD-style energy + forces on gfx1250 (MI455X).
// All matmuls run on the CDNA5 matrix pipes via V_WMMA_F32_16X16X4_F32,
// batching 16 pairs/atoms per wave32 as the M dimension of a WMMA tile.
// Weights are pre-packed into zero-padded LDS images (fwd + transposed bwd)
// and all hot loops are fully branch-free: unconditional LDS loads, select-
// stores, A-fragments held in registers across N-tiles, WMMAs back-to-back.
// Only the per-atom descriptor D and dE/dD (16MB) round-trip global memory
// (L2-resident at 192MB).
// ---------------------------------------------------------------------------

namespace {

constexpr int   kNS    = 8;
constexpr int   kNP    = 4096;
constexpr int   kMN    = 32;
constexpr int   kNAtom = kNS * kNP;        // 32768
constexpr int   kNPair = kNAtom * kMN;     // 1048576
constexpr float kL     = 64.0f;

constexpr int ceil4c(int x)  { return (x + 3) & ~3; }
constexpr int ceil16c(int x) { return (x + 15) & ~15; }

// pair pyramid dims 1/2 -> 2 -> 4 -> 8 -> 16 -> 32
constexpr int DD[5]     = {2, 4, 8, 16, 32};
constexpr int AOFF[5]   = {0, 4, 8, 16, 32};     // act col offsets (row width 64)
constexpr int PF_OFF[5] = {0, 64, 128, 192, 320};// fwd B-image offsets, total 832
constexpr int PB_OFF[5] = {0, 64, 128, 256, 512};// bwd BT-image offsets, total 1024
constexpr int BOFF[5]   = {0, 16, 32, 48, 64};   // padded bias offsets, total 96

// fit net dims 64 -> 16 -> 8 -> 4 -> 2 -> 1
constexpr int FD[5]     = {16, 8, 4, 2, 1};
constexpr int FAOFF[5]  = {0, 16, 24, 28, 32};         // act col offsets
constexpr int FF_OFF[5] = {0, 1024, 1280, 1408, 1472}; // fwd images, total 1536
constexpr int FB_OFF[5] = {0, 1024, 1152, 1216, 1280}; // bwd images, total 1344
constexpr int FBOFF[5]  = {0, 16, 32, 48, 64};         // padded bias, total 80

// bank-conflict-free, 8B-aligned LDS row strides
constexpr int SA  = 68;  // pair activation rows (max col 64)
constexpr int SG  = 36;  // pair gradient rows  (max col 32)
constexpr int SD  = 66;  // fit descriptor rows (max col 64)
constexpr int SFA = 52;  // fit activation rows (max col 48 incl 16-wide stores)
constexpr int SGA = 66;  // fit gradient rows, wide (max col 64)
constexpr int SGB = 20;  // fit gradient rows, narrow (max col 16)

typedef __attribute__((ext_vector_type(2))) float v2f;
typedef __attribute__((ext_vector_type(8))) float v8f;

struct KParams {
  const float* R;             // positions (NS*NP, 2)
  const int*   NL;            // neighbor list (NS*NP, MN)
  const float* p1W[5]; const float* p1b[5];
  const float* p2W[5]; const float* p2b[5];
  const float* fW[5];  const float* fb[5];
  const float* lW;     const float* lb;
  float* E;                   // d_out[0..7]
  float* F;                   // d_out + 8, forces (NS*NP, 2)
  float* D;                   // ws: per-atom descriptor (NAtom, 64)
  float* GD;                  // ws: dE/dD            (NAtom, 64)
};

// branch-free tanh: v_exp_f32 + v_rcp_f32, clamped so e stays finite
__device__ inline float fast_tanh(float x) {
  x = fminf(15.0f, fmaxf(-15.0f, x));
  const float e = __builtin_amdgcn_exp2f(x * 2.885390081777927f); // exp(2x)
  const float r = __builtin_amdgcn_rcpf(e + 1.0f);
  return (e - 1.0f) * r;
}

// ---- weight packing into zero-padded LDS images ---------------------------
__device__ inline void pack_fwd(float* dst, const float* src, int K, int N) {
  const int KP = ceil4c(K), NP = ceil16c(N);
  for (int i = threadIdx.x; i < KP * NP; i += blockDim.x) {
    const int k = i / NP, n = i % NP;
    dst[i] = (k < K && n < N) ? src[k * N + n] : 0.0f;
  }
}
__device__ inline void pack_bwd(float* dst, const float* src, int K, int N) {
  const int NR = ceil4c(N), KP = ceil16c(K);
  for (int i = threadIdx.x; i < NR * KP; i += blockDim.x) {
    const int r = i / KP, c = i % KP;
    dst[i] = (r < N && c < K) ? src[c * N + r] : 0.0f;
  }
}
__device__ inline void pack_bias(float* dst, const float* src, int N) {
  const int NP = ceil16c(N);
  for (int i = threadIdx.x; i < NP; i += blockDim.x)
    dst[i] = (i < N) ? src[i] : 0.0f;
}

// ---------------------------------------------------------------------------
// acc[t] += X(16 x KP) * Bimg[:, t*16 .. t*16+15]  via V_WMMA_F32_16X16X4_F32.
// A fragment is loaded once per k-step and reused (registers) for all tiles;
// the NT WMMAs per k-step issue back-to-back.
// ---------------------------------------------------------------------------
template <int KP, int NPI, int NT>
__device__ inline void mm_tiles(const float* __restrict__ xb, int sx,
                                const float* __restrict__ Bimg,
                                v8f (&acc)[NT], int lane) {
  const int lg = lane >> 4, lr = lane & 15;
#pragma unroll
  for (int k0 = 0; k0 < KP; k0 += 4) {
    const int ka = k0 + 2 * lg;
    v2f a;
    a.x = xb[lr * sx + ka];
    a.y = xb[lr * sx + ka + 1];
#pragma unroll
    for (int t = 0; t < NT; ++t) {
      v2f b;
      b.x = Bimg[ka * NPI + t * 16 + lr];
      b.y = Bimg[(ka + 1) * NPI + t * 16 + lr];
      acc[t] = __builtin_amdgcn_wmma_f32_16x16x4_f32(false, a, false, b,
                                                     (short)0, acc[t], false, false);
    }
  }
}

// ---------------------------------------------------------------------------
// Forward layer: ob = combine(tanh(xb @ W + b), xb)
//   MODE 0: ob = h     MODE 1 (N == 2K, K pow2): ob = concat(x,x) + h
// Stores the full 16-wide tile(s) with a register select (pad cols get 0);
// pad stores only land in later layers' not-yet-written regions.
// ---------------------------------------------------------------------------
template <int K, int N, int MODE>
__device__ inline void mlp_fwd(const float* __restrict__ xb, int sx,
                               float* __restrict__ ob, int so,
                               const float* __restrict__ Bimg,
                               const float* __restrict__ bl, int lane) {
  constexpr int KP  = ceil4c(K);
  constexpr int NPI = ceil16c(N);
  constexpr int NT  = NPI / 16;
  const int lg = lane >> 4, lr = lane & 15;
  v8f acc[NT] = {};
  mm_tiles<KP, NPI, NT>(xb, sx, Bimg, acc, lane);
#pragma unroll
  for (int t = 0; t < NT; ++t) {
    const int col = t * 16 + lr;
#pragma unroll
    for (int r = 0; r < 8; ++r) {
      const int row = r + 8 * lg;
      float o = fast_tanh(acc[t][r] + bl[col]);
      if (MODE == 1) o += xb[row * sx + (col & (K - 1))];
      ob[row * so + col] = (col < N) ? o : 0.0f;
    }
  }
  __syncthreads();
}

// ---------------------------------------------------------------------------
// Backward layer. gIn: 16 x N upstream grads (cols [N,16) zero, DESTROYED
// -> gpre). Writes gOut (16 x K) = concat-part + gpre @ W^T with select-
// stores (pad cols zeroed). MODE 1: N == 2K, K <= 16. Fully branch-free.
// ---------------------------------------------------------------------------
template <int N, int K, int MODE>
__device__ inline void mlp_bwd(float* __restrict__ gIn, int sgin,
                               const float* __restrict__ xprev, int sxp,
                               const float* __restrict__ xout, int sxo,
                               float* __restrict__ gOut, int sgo,
                               const float* __restrict__ BT, int lane) {
  constexpr int NP4 = ceil4c(N);
  constexpr int KPI = ceil16c(K);
  constexpr int NT  = KPI / 16;
  const int lg = lane >> 4, lr = lane & 15;
  // capture concat contribution before gIn is overwritten (lanes with
  // lr >= K read in-range garbage that the select-store discards later)
  v8f cc = {};
  if (MODE == 1) {
#pragma unroll
    for (int r = 0; r < 8; ++r) {
      const int row = r + 8 * lg;
      cc[r] = gIn[row * sgin + lr] + gIn[row * sgin + lr + K];
    }
    __syncthreads();
  }
  // gIn <- gpre = gIn * (1 - h^2); full 16-wide tiles, pad cols stay zero
#pragma unroll
  for (int n0 = 0; n0 < ceil16c(NP4); n0 += 16) {
    const int col = n0 + lr;
#pragma unroll
    for (int r = 0; r < 8; ++r) {
      const int row = r + 8 * lg;
      float h = xout[row * sxo + col];
      if (MODE == 1) h -= xprev[row * sxp + (col & (K - 1))];
      gIn[row * sgin + col] *= (1.0f - h * h);
    }
  }
  __syncthreads();
  // gOut = concat-part + gpre @ W^T
  v8f acc[NT] = {};
  if (MODE == 1) acc[0] = cc;
  mm_tiles<NP4, KPI, NT>(gIn, sgin, BT, acc, lane);
#pragma unroll
  for (int t = 0; t < NT; ++t) {
    const int col = t * 16 + lr;
#pragma unroll
    for (int r = 0; r < 8; ++r)
      gOut[(r + 8 * lg) * sgo + col] = (col < K) ? acc[t][r] : 0.0f;
  }
  __syncthreads();
}

// ---------------------------------------------------------------------------
__global__ void k_zero(float* a, int n) {
  int i = blockIdx.x * blockDim.x + threadIdx.x;
  if (i < n) a[i] = 0.0f;
}

// ---------------------------------------------------------------------------
// Kernel 1: pair forward.  4 waves/block, one wave = 16 pairs.
// ---------------------------------------------------------------------------
constexpr int WPB1 = 4, TPB1 = WPB1 * 32;

__global__ __launch_bounds__(TPB1) void k_pair_fwd(KParams P) {
  __shared__ float sP1F[832], sP2F[832], sB1[96], sB2[96];
  __shared__ float sAct1[WPB1][16 * SA], sAct2[WPB1][16 * SA];
  __shared__ float sIn1[WPB1][16 * 4], sIn2[WPB1][16 * 4];

  for (int l = 0; l < 5; ++l) {
    pack_fwd(sP1F + PF_OFF[l], P.p1W[l], l ? DD[l - 1] : 1, DD[l]);
    pack_fwd(sP2F + PF_OFF[l], P.p2W[l], l ? DD[l - 1] : 2, DD[l]);
    pack_bias(sB1 + BOFF[l], P.p1b[l], DD[l]);
    pack_bias(sB2 + BOFF[l], P.p2b[l], DD[l]);
  }
  __syncthreads();

  const int wid = threadIdx.x >> 5, lane = threadIdx.x & 31;
  const int tile   = blockIdx.x * WPB1 + wid;
  const int atom   = tile >> 1;           // 32 neighbors = 2 tiles per atom
  const int base   = tile << 4;
  const int sample = atom >> 12;          // NP = 4096

  if (lane < 16) {
    const int nb = P.NL[base + lane];
    const float m = (nb >= 0) ? 1.0f : 0.0f;
    const int aj  = (sample << 12) + (nb >= 0 ? nb : 0);
    float dx = P.R[aj * 2 + 0] - P.R[atom * 2 + 0];
    float dy = P.R[aj * 2 + 1] - P.R[atom * 2 + 1];
    dx -= kL * rintf(dx * (1.0f / kL));
    dy -= kL * rintf(dy * (1.0f / kL));
    const float d2 = m * (dx * dx + dy * dy) + (1.0f - m);
    const float ri = 1.0f / sqrtf(d2);
    sIn1[wid][lane * 4 + 0] = m * ri;
    sIn1[wid][lane * 4 + 1] = 0.0f; sIn1[wid][lane * 4 + 2] = 0.0f; sIn1[wid][lane * 4 + 3] = 0.0f;
    sIn2[wid][lane * 4 + 0] = m * dx * ri * ri;
    sIn2[wid][lane * 4 + 1] = m * dy * ri * ri;
    sIn2[wid][lane * 4 + 2] = 0.0f; sIn2[wid][lane * 4 + 3] = 0.0f;
  }
  __syncthreads();

  float* A1 = sAct1[wid];
  float* A2 = sAct2[wid];
  mlp_fwd<1, 2, 0>(sIn1[wid], 4, A1 + AOFF[0], SA, sP1F + PF_OFF[0], sB1 + BOFF[0], lane);
  mlp_fwd<2, 2, 0>(sIn2[wid], 4, A2 + AOFF[0], SA, sP2F + PF_OFF[0], sB2 + BOFF[0], lane);
  mlp_fwd<2, 4, 1>(A1 + AOFF[0], SA, A1 + AOFF[1], SA, sP1F + PF_OFF[1], sB1 + BOFF[1], lane);
  mlp_fwd<2, 4, 1>(A2 + AOFF[0], SA, A2 + AOFF[1], SA, sP2F + PF_OFF[1], sB2 + BOFF[1], lane);
  mlp_fwd<4, 8, 1>(A1 + AOFF[1], SA, A1 + AOFF[2], SA, sP1F + PF_OFF[2], sB1 + BOFF[2], lane);
  mlp_fwd<4, 8, 1>(A2 + AOFF[1], SA, A2 + AOFF[2], SA, sP2F + PF_OFF[2], sB2 + BOFF[2], lane);
  mlp_fwd<8, 16, 1>(A1 + AOFF[2], SA, A1 + AOFF[3], SA, sP1F + PF_OFF[3], sB1 + BOFF[3], lane);
  mlp_fwd<8, 16, 1>(A2 + AOFF[2], SA, A2 + AOFF[3], SA, sP2F + PF_OFF[3], sB2 + BOFF[3], lane);
  mlp_fwd<16, 32, 1>(A1 + AOFF[3], SA, A1 + AOFF[4], SA, sP1F + PF_OFF[4], sB1 + BOFF[4], lane);
  mlp_fwd<16, 32, 1>(A2 + AOFF[3], SA, A2 + AOFF[4], SA, sP2F + PF_OFF[4], sB2 + BOFF[4], lane);

  // D[atom][c] += sum_rows out[row][c]*s[row]; each lane owns 2 columns
  const float* O1 = A1 + AOFF[4];
  const float* O2 = A2 + AOFF[4];
  float acc1 = 0.0f, acc2 = 0.0f;
  for (int r = 0; r < 16; ++r) {
    const float s = sIn1[wid][r * 4];
    acc1 += O1[r * SA + lane] * s;
    acc2 += O2[r * SA + lane] * s;
  }
  atomicAdd(&P.D[atom * 64 + lane], acc1);
  atomicAdd(&P.D[atom * 64 + 32 + lane], acc2);
}

// ---------------------------------------------------------------------------
// Kernel 2: fit net forward + backward.  4 waves/block, one wave = 16 atoms.
// ---------------------------------------------------------------------------
constexpr int WPB2 = 4, TPB2 = WPB2 * 32;

__global__ __launch_bounds__(TPB2) void k_fit(KParams P) {
  __shared__ float sFF[1536], sFB[1344], sFb[80];
  __shared__ float sD[WPB2][16 * SD], sAct[WPB2][16 * SFA];
  __shared__ float sGA[WPB2][16 * SGA], sGB[WPB2][16 * SGB];

  for (int l = 0; l < 5; ++l) {
    pack_fwd(sFF + FF_OFF[l], P.fW[l], l ? FD[l - 1] : 64, FD[l]);
    pack_bwd(sFB + FB_OFF[l], P.fW[l], l ? FD[l - 1] : 64, FD[l]);
    pack_bias(sFb + FBOFF[l], P.fb[l], FD[l]);
  }
  __syncthreads();

  const int wid = threadIdx.x >> 5, lane = threadIdx.x & 31;
  const int tile = blockIdx.x * WPB2 + wid;
  const int a0   = tile << 4;

  for (int i = lane; i < 16 * 64; i += 32) {
    const int row = i >> 6, c = i & 63;
    sD[wid][row * SD + c] = P.D[a0 * 64 + i];
  }
  __syncthreads();

  float* A = sAct[wid];
  mlp_fwd<64, 16, 0>(sD[wid], SD, A + FAOFF[0], SFA, sFF + FF_OFF[0], sFb + FBOFF[0], lane);
  mlp_fwd<16, 8, 0>(A + FAOFF[0], SFA, A + FAOFF[1], SFA, sFF + FF_OFF[1], sFb + FBOFF[1], lane);
  mlp_fwd<8, 4, 0>(A + FAOFF[1], SFA, A + FAOFF[2], SFA, sFF + FF_OFF[2], sFb + FBOFF[2], lane);
  mlp_fwd<4, 2, 0>(A + FAOFF[2], SFA, A + FAOFF[3], SFA, sFF + FF_OFF[3], sFb + FBOFF[3], lane);
  mlp_fwd<2, 1, 0>(A + FAOFF[3], SFA, A + FAOFF[4], SFA, sFF + FF_OFF[4], sFb + FBOFF[4], lane);

  const float lw = P.lW[0], lb = P.lb[0];
  if (lane < 16) {
    const float Fv = A[lane * SFA + FAOFF[4]] * lw + lb;
    atomicAdd(&P.E[(a0 + lane) >> 12], Fv);
    sGB[wid][lane * SGB + 0] = lw;                    // seed dE/dx5 = lW
    sGB[wid][lane * SGB + 1] = 0.0f;
    sGB[wid][lane * SGB + 2] = 0.0f;
    sGB[wid][lane * SGB + 3] = 0.0f;
  }
  __syncthreads();

  mlp_bwd<1, 2, 0>(sGB[wid], SGB, A + FAOFF[3], SFA, A + FAOFF[4], SFA, sGA[wid], SGA, sFB + FB_OFF[4], lane);
  mlp_bwd<2, 4, 0>(sGA[wid], SGA, A + FAOFF[2], SFA, A + FAOFF[3], SFA, sGB[wid], SGB, sFB + FB_OFF[3], lane);
  mlp_bwd<4, 8, 0>(sGB[wid], SGB, A + FAOFF[1], SFA, A + FAOFF[2], SFA, sGA[wid], SGA, sFB + FB_OFF[2], lane);
  mlp_bwd<8, 16, 0>(sGA[wid], SGA, A + FAOFF[0], SFA, A + FAOFF[1], SFA, sGB[wid], SGB, sFB + FB_OFF[1], lane);
  mlp_bwd<16, 64, 0>(sGB[wid], SGB, sD[wid], SD, A + FAOFF[0], SFA, sGA[wid], SGA, sFB + FB_OFF[0], lane);

  for (int i = lane; i < 16 * 64; i += 32) {
    const int row = i >> 6, c = i & 63;
    P.GD[a0 * 64 + i] = sGA[wid][row * SGA + c];
  }
}

// ---------------------------------------------------------------------------
// Kernel 3: pair backward (recompute fwd, backprop, scatter forces).
// 2 waves/block (LDS budget), one wave = 16 pairs.
// ---------------------------------------------------------------------------
constexpr int WPB3 = 2, TPB3 = WPB3 * 32;

__global__ __launch_bounds__(TPB3) void k_pair_bwd(KParams P) {
  __shared__ float sP1F[832], sP2F[832], sP1B[1024], sP2B[1024], sB1[96], sB2[96];
  __shared__ float sAct1[WPB3][16 * SA], sAct2[WPB3][16 * SA];
  __shared__ float sIn1[WPB3][16 * 4], sIn2[WPB3][16 * 4];
  __shared__ float sScal[WPB3][16 * 8];
  __shared__ float sGDw[WPB3][64];
  __shared__ float sGA[WPB3][16 * SG], sGB[WPB3][16 * SG];

  for (int l = 0; l < 5; ++l) {
    pack_fwd(sP1F + PF_OFF[l], P.p1W[l], l ? DD[l - 1] : 1, DD[l]);
    pack_fwd(sP2F + PF_OFF[l], P.p2W[l], l ? DD[l - 1] : 2, DD[l]);
    pack_bwd(sP1B + PB_OFF[l], P.p1W[l], l ? DD[l - 1] : 1, DD[l]);
    pack_bwd(sP2B + PB_OFF[l], P.p2W[l], l ? DD[l - 1] : 2, DD[l]);
    pack_bias(sB1 + BOFF[l], P.p1b[l], DD[l]);
    pack_bias(sB2 + BOFF[l], P.p2b[l], DD[l]);
  }
  __syncthreads();

  const int wid = threadIdx.x >> 5, lane = threadIdx.x & 31;
  const int lg = lane >> 4, lr = lane & 15;
  const int tile   = blockIdx.x * WPB3 + wid;
  const int atom   = tile >> 1;
  const int base   = tile << 4;
  const int sample = atom >> 12;

  if (lane < 16) {
    const int nb = P.NL[base + lane];
    const float m = (nb >= 0) ? 1.0f : 0.0f;
    const int aj  = (sample << 12) + (nb >= 0 ? nb : 0);
    float dx = P.R[aj * 2 + 0] - P.R[atom * 2 + 0];
    float dy = P.R[aj * 2 + 1] - P.R[atom * 2 + 1];
    dx -= kL * rintf(dx * (1.0f / kL));
    dy -= kL * rintf(dy * (1.0f / kL));
    const float d2 = m * (dx * dx + dy * dy) + (1.0f - m);
    const float ri = 1.0f / sqrtf(d2);
    sIn1[wid][lane * 4 + 0] = m * ri;
    sIn1[wid][lane * 4 + 1] = 0.0f; sIn1[wid][lane * 4 + 2] = 0.0f; sIn1[wid][lane * 4 + 3] = 0.0f;
    sIn2[wid][lane * 4 + 0] = m * dx * ri * ri;
    sIn2[wid][lane * 4 + 1] = m * dy * ri * ri;
    sIn2[wid][lane * 4 + 2] = 0.0f; sIn2[wid][lane * 4 + 3] = 0.0f;
    sScal[wid][lane * 8 + 0] = m;
    sScal[wid][lane * 8 + 1] = dx;
    sScal[wid][lane * 8 + 2] = dy;
    sScal[wid][lane * 8 + 3] = ri;
  }
  for (int c = lane; c < 64; c += 32) sGDw[wid][c] = P.GD[atom * 64 + c];
  __syncthreads();

  // ---- recompute forward activations ----
  float* A1 = sAct1[wid];
  float* A2 = sAct2[wid];
  mlp_fwd<1, 2, 0>(sIn1[wid], 4, A1 + AOFF[0], SA, sP1F + PF_OFF[0], sB1 + BOFF[0], lane);
  mlp_fwd<2, 2, 0>(sIn2[wid], 4, A2 + AOFF[0], SA, sP2F + PF_OFF[0], sB2 + BOFF[0], lane);
  mlp_fwd<2, 4, 1>(A1 + AOFF[0], SA, A1 + AOFF[1], SA, sP1F + PF_OFF[1], sB1 + BOFF[1], lane);
  mlp_fwd<2, 4, 1>(A2 + AOFF[0], SA, A2 + AOFF[1], SA, sP2F + PF_OFF[1], sB2 + BOFF[1], lane);
  mlp_fwd<4, 8, 1>(A1 + AOFF[1], SA, A1 + AOFF[2], SA, sP1F + PF_OFF[2], sB1 + BOFF[2], lane);
  mlp_fwd<4, 8, 1>(A2 + AOFF[1], SA, A2 + AOFF[2], SA, sP2F + PF_OFF[2], sB2 + BOFF[2], lane);
  mlp_fwd<8, 16, 1>(A1 + AOFF[2], SA, A1 + AOFF[3], SA, sP1F + PF_OFF[3], sB1 + BOFF[3], lane);
  mlp_fwd<8, 16, 1>(A2 + AOFF[2], SA, A2 + AOFF[3], SA, sP2F + PF_OFF[3], sB2 + BOFF[3], lane);
  mlp_fwd<16, 32, 1>(A1 + AOFF[3], SA, A1 + AOFF[4], SA, sP1F + PF_OFF[4], sB1 + BOFF[4], lane);
  mlp_fwd<16, 32, 1>(A2 + AOFF[3], SA, A2 + AOFF[4], SA, sP2F + PF_OFF[4], sB2 + BOFF[4], lane);

  // ---- backprop p1 : upstream dE/dA1 = GD[0:32] * s ----
  for (int n0 = 0; n0 < 32; n0 += 16) {
    const int col = n0 + lr;
#pragma unroll
    for (int r = 0; r < 8; ++r) {
      const int row = r + 8 * lg;
      sGA[wid][row * SG + col] = sGDw[wid][col] * sIn1[wid][row * 4];
    }
  }
  __syncthreads();
  mlp_bwd<32, 16, 1>(sGA[wid], SG, A1 + AOFF[3], SA, A1 + AOFF[4], SA, sGB[wid], SG, sP1B + PB_OFF[4], lane);
  mlp_bwd<16, 8, 1>(sGB[wid], SG, A1 + AOFF[2], SA, A1 + AOFF[3], SA, sGA[wid], SG, sP1B + PB_OFF[3], lane);
  mlp_bwd<8, 4, 1>(sGA[wid], SG, A1 + AOFF[1], SA, A1 + AOFF[2], SA, sGB[wid], SG, sP1B + PB_OFF[2], lane);
  mlp_bwd<4, 2, 1>(sGB[wid], SG, A1 + AOFF[0], SA, A1 + AOFF[1], SA, sGA[wid], SG, sP1B + PB_OFF[1], lane);
  mlp_bwd<2, 1, 0>(sGA[wid], SG, sIn1[wid], 4, A1 + AOFF[0], SA, sGB[wid], SG, sP1B + PB_OFF[0], lane);
  if (lane < 16) sScal[wid][lane * 8 + 4] = sGB[wid][lane * SG + 0];  // dE/ds via p1
  __syncthreads();

  // ---- backprop p2 : upstream dE/dA2 = GD[32:64] * s ----
  for (int n0 = 0; n0 < 32; n0 += 16) {
    const int col = n0 + lr;
#pragma unroll
    for (int r = 0; r < 8; ++r) {
      const int row = r + 8 * lg;
      sGA[wid][row * SG + col] = sGDw[wid][32 + col] * sIn1[wid][row * 4];
    }
  }
  __syncthreads();
  mlp_bwd<32, 16, 1>(sGA[wid], SG, A2 + AOFF[3], SA, A2 + AOFF[4], SA, sGB[wid], SG, sP2B + PB_OFF[4], lane);
  mlp_bwd<16, 8, 1>(sGB[wid], SG, A2 + AOFF[2], SA, A2 + AOFF[3], SA, sGA[wid], SG, sP2B + PB_OFF[3], lane);
  mlp_bwd<8, 4, 1>(sGA[wid], SG, A2 + AOFF[1], SA, A2 + AOFF[2], SA, sGB[wid], SG, sP2B + PB_OFF[2], lane);
  mlp_bwd<4, 2, 1>(sGB[wid], SG, A2 + AOFF[0], SA, A2 + AOFF[1], SA, sGA[wid], SG, sP2B + PB_OFF[1], lane);
  mlp_bwd<2, 2, 0>(sGA[wid], SG, sIn2[wid], 4, A2 + AOFF[0], SA, sGB[wid], SG, sP2B + PB_OFF[0], lane);

  // ---- chain rule to positions, scatter forces ----
  if (lane < 16) {
    const int row = lane;
    const float m  = sScal[wid][row * 8 + 0];
    const float dx = sScal[wid][row * 8 + 1];
    const float dy = sScal[wid][row * 8 + 2];
    const float ri = sScal[wid][row * 8 + 3];
    float gs = sScal[wid][row * 8 + 4];
    const float* O1 = A1 + AOFF[4];
    const float* O2 = A2 + AOFF[4];
    float gdir = 0.0f;                      // product-rule term of L = pyr * s
    for (int c = 0; c < 32; ++c)
      gdir += sGDw[wid][c] * O1[row * SA + c] + sGDw[wid][32 + c] * O2[row * SA + c];
    gs += gdir;
    const float gxc = sGB[wid][row * SG + 0];   // dE/d(dx*rinv^2)
    const float gyc = sGB[wid][row * SG + 1];
    const float ri2 = ri * ri, ri3 = ri2 * ri, ri4 = ri2 * ri2;
    const float gd_x = m * (-gs * dx * ri3 + gxc * (ri2 - 2.0f * dx * dx * ri4)
                            - 2.0f * gyc * dx * dy * ri4);
    const float gd_y = m * (-gs * dy * ri3 + gyc * (ri2 - 2.0f * dy * dy * ri4)
                            - 2.0f * gxc * dx * dy * ri4);
    // E depends on d = Rj - Ri ; F = -dE/dR
    atomicAdd(&P.F[atom * 2 + 0], gd_x);
    atomicAdd(&P.F[atom * 2 + 1], gd_y);
    const int nb = P.NL[base + row];
    if (nb >= 0) {
      const int aj = (sample << 12) + nb;
      atomicAdd(&P.F[aj * 2 + 0], -gd_x);
      atomicAdd(&P.F[aj * 2 + 1], -gd_y);
    }
  }
}

}  // namespace

// ---------------------------------------------------------------------------
extern "C" void kernel_launch(void* const* d_in, const int* in_sizes, int n_in,
                              void* d_out, int out_size, void* d_ws, size_t ws_size,
                              hipStream_t stream) {
  (void)in_sizes; (void)n_in; (void)ws_size;
  KParams P;
  P.R  = (const float*)d_in[0];
  P.NL = (const int*)d_in[1];
  for (int i = 0; i < 5; ++i) {
    P.p1W[i] = (const float*)d_in[2 + i];
    P.p1b[i] = (const float*)d_in[7 + i];
    P.p2W[i] = (const float*)d_in[12 + i];
    P.p2b[i] = (const float*)d_in[17 + i];
    P.fW[i]  = (const float*)d_in[22 + i];
    P.fb[i]  = (const float*)d_in[27 + i];
  }
  P.lW = (const float*)d_in[32];
  P.lb = (const float*)d_in[33];

  float* out = (float*)d_out;
  P.E = out;
  P.F = out + kNS;
  float* ws = (float*)d_ws;
  P.D  = ws;
  P.GD = ws + (size_t)kNAtom * 64;

  const int nz1 = out_size;        // E + forces
  const int nz2 = kNAtom * 64;     // descriptor accumulator
  hipLaunchKernelGGL(k_zero, dim3((nz1 + 255) / 256), dim3(256), 0, stream, out, nz1);
  hipLaunchKernelGGL(k_zero, dim3((nz2 + 255) / 256), dim3(256), 0, stream, P.D, nz2);

  hipLaunchKernelGGL(k_pair_fwd, dim3(kNPair / 16 / WPB1), dim3(TPB1), 0, stream, P);
  hipLaunchKernelGGL(k_fit,      dim3(kNAtom / 16 / WPB2), dim3(TPB2), 0, stream, P);
  hipLaunchKernelGGL(k_pair_bwd, dim3(kNPair / 16 / WPB3), dim3(TPB3), 0, stream, P);
}